// MultiHeadAttention_66228395704385
// MI455X (gfx1250) — compile-verified
//
#include <hip/hip_runtime.h>
#include <hip/hip_bf16.h>
#include <cstddef>

// ---------------- problem constants ----------------
#define BATCH   4
#define NSEQ    2048
#define DMODEL  512
#define HEADS   8
#define DH      64
#define INNER   (HEADS * DH)      // 512
#define QKVW    (3 * INNER)       // 1536
#define ROWS    (BATCH * NSEQ)    // 8192
#define SCALE   0.125f            // 1/sqrt(64)
#define INV_TAU (1.0f / 50.0f)
#define LN_EPS  1e-5f

typedef float v2f __attribute__((ext_vector_type(2)));
typedef float v8f __attribute__((ext_vector_type(8)));

// ---- CDNA5 async LDS<->global path (guarded; falls back cleanly) ----
// clang's builtin signature (from its own diagnostic):
//   (int4 addrspace(1)*, int4 addrspace(3)*, imm offset, imm cpol)
#if __has_builtin(__builtin_amdgcn_global_load_async_to_lds_b128) && \
    __has_builtin(__builtin_amdgcn_global_store_async_from_lds_b128)
#define HAVE_ASYNC_LDS 1
typedef int v4i __attribute__((vector_size(16)));
typedef __attribute__((address_space(1))) v4i* gp4_t;
typedef __attribute__((address_space(3))) v4i* lp4_t;
#define ASYNC_G2L(gsrc, ldst)                                                  \
  __builtin_amdgcn_global_load_async_to_lds_b128((gp4_t)(gsrc), (lp4_t)(ldst), \
                                                 0, 0)
#define ASYNC_L2G(gdst, lsrc)                                                  \
  __builtin_amdgcn_global_store_async_from_lds_b128((gp4_t)(gdst),             \
                                                    (lp4_t)(lsrc), 0, 0)
#else
#define HAVE_ASYNC_LDS 0
#endif

__device__ __forceinline__ void wait_async0() {
#if __has_builtin(__builtin_amdgcn_s_wait_asynccnt)
  __builtin_amdgcn_s_wait_asynccnt(0);
#else
  asm volatile("s_wait_asynccnt 0" ::: "memory");
#endif
}

// D = A(16x4 f32) * B(4x16 f32) + C(16x16 f32)   -- V_WMMA_F32_16X16X4_F32
__device__ __forceinline__ v8f wmma4(v2f a, v2f b, v8f c) {
  return __builtin_amdgcn_wmma_f32_16x16x4_f32(
      /*neg_a=*/false, a, /*neg_b=*/false, b,
      /*c_mod=*/(short)0, c, /*reuse_a=*/false, /*reuse_b=*/false);
}

// =====================================================================
// Kernel 1: qkv = x @ w_qkv     [8192 x 512] * [512 x 1536]
// Block = 256 thr (8 waves). Block computes 16 rows x 512 cols.
// grid = (512 rowtiles, 3 colblocks)
// =====================================================================
__global__ void __launch_bounds__(256)
qkv_gemm_kernel(const float* __restrict__ x, const float* __restrict__ w,
                float* __restrict__ qkv) {
  __shared__ float Xs[16][DMODEL + 4];
  const int row0 = blockIdx.x * 16;
  const int col0 = blockIdx.y * 512;
  const int tid  = threadIdx.x;
#if HAVE_ASYNC_LDS
  for (int i = tid; i < (16 * DMODEL) / 4; i += 256) {
    int c4 = i * 4;
    int r = c4 >> 9, c = c4 & (DMODEL - 1);
    ASYNC_G2L(x + (size_t)(row0 + r) * DMODEL + c, &Xs[r][c]);
  }
  wait_async0();
#else
  for (int i = tid; i < 16 * DMODEL; i += 256) {
    int r = i >> 9, c = i & (DMODEL - 1);
    Xs[r][c] = x[(size_t)(row0 + r) * DMODEL + c];
  }
#endif
  __syncthreads();

  const int wave = tid >> 5, lane = tid & 31;
  const int half = lane >> 4, l16 = lane & 15, koff = half * 2;
  const int wcol0 = col0 + wave * 64;

  v8f acc[4] = {};
  for (int kk = 0; kk < DMODEL; kk += 4) {
    v2f a;
    a.x = Xs[l16][kk + koff];
    a.y = Xs[l16][kk + koff + 1];
#pragma unroll
    for (int t = 0; t < 4; ++t) {
      int col = wcol0 + t * 16 + l16;
      v2f b;
      b.x = w[(size_t)(kk + koff) * QKVW + col];
      b.y = w[(size_t)(kk + koff + 1) * QKVW + col];
      acc[t] = wmma4(a, b, acc[t]);
    }
  }
#pragma unroll
  for (int t = 0; t < 4; ++t)
#pragma unroll
    for (int j = 0; j < 8; ++j) {
      int r = half ? (j + 8) : j;
      qkv[(size_t)(row0 + r) * QKVW + wcol0 + t * 16 + l16] = acc[t][j];
    }
}

// =====================================================================
// Kernel 2: full attention for one (b, h, 16-query tile).
// Block = 256 thr (8 waves); score slab [16][2052] lives in LDS (~131KB).
// grid = 4 * 8 * 128 = 4096
// =====================================================================
#define SS (NSEQ + 4)   // padded score stride (2052; 2052 % 64 != 0 -> bank skew)

__global__ void __launch_bounds__(256)
attention_kernel(const float* __restrict__ qkv,
                 float* __restrict__ attn,   // [4,8,2048,2048]
                 float* __restrict__ aout) { // [4,2048,512] (b,q,h*64+f)
  extern __shared__ float smem[];
  float* Sc   = smem;                 // [16][SS]
  float* Qs   = Sc + 16 * SS;         // [16][64]
  float* Red  = Qs + 16 * 64;         // [16][16]
  float* RowM = Red + 256;            // [16]
  float* RowS = RowM + 16;            // [16]

  const int blk = blockIdx.x;
  const int qt = blk & 127, h = (blk >> 7) & 7, b = blk >> 10;
  const int q0 = qt * 16;
  const int tid = threadIdx.x;
  const int wave = tid >> 5, lane = tid & 31;
  const int half = lane >> 4, l16 = lane & 15, koff = half * 2;

  const size_t qkv_b = (size_t)b * NSEQ * QKVW;
  // stage Q tile: Qs[r][f] = Q[b, q0+r, h, f]
#if HAVE_ASYNC_LDS
  {
    int c4 = tid * 4;                      // exactly 256 16B chunks
    int r = c4 >> 6, f = c4 & 63;
    ASYNC_G2L(qkv + qkv_b + (size_t)(q0 + r) * QKVW + h * DH + f,
              Qs + r * DH + f);
    wait_async0();
  }
#else
  for (int i = tid; i < 16 * DH; i += 256) {
    int r = i >> 6, f = i & 63;
    Qs[i] = qkv[qkv_b + (size_t)(q0 + r) * QKVW + h * DH + f];
  }
#endif
  __syncthreads();

  // ---- stage 2: scores. wave owns keys [wave*256, wave*256+256).
  // Two subtiles in flight -> independent WMMA chains + shared A-frag.
  const float* kbase = qkv + qkv_b + INNER + h * DH;      // K[key][f]
  for (int st = 0; st < 16; st += 2) {
    const int key0 = wave * 256 + st * 16;
    v8f c0 = {}, c1 = {};
    const float* kr0 = kbase + (size_t)(key0 + l16) * QKVW;
    const float* kr1 = kr0 + (size_t)16 * QKVW;
    for (int kk = 0; kk < DH; kk += 4) {
      v2f a; a.x = Qs[l16 * DH + kk + koff]; a.y = Qs[l16 * DH + kk + koff + 1];
      v2f b0; b0.x = kr0[kk + koff]; b0.y = kr0[kk + koff + 1];
      v2f b1; b1.x = kr1[kk + koff]; b1.y = kr1[kk + koff + 1];
      c0 = wmma4(a, b0, c0);
      c1 = wmma4(a, b1, c1);
    }
    const int kcol = key0 + l16;
    const float dec0 = __expf(-(float)kcol * INV_TAU);
    const float dec1 = __expf(-(float)(kcol + 16) * INV_TAU);
#pragma unroll
    for (int j = 0; j < 8; ++j) {
      int r = half ? (j + 8) : j;
      Sc[r * SS + kcol]      = c0[j] * SCALE + dec0;
      Sc[r * SS + kcol + 16] = c1[j] * SCALE + dec1;
    }
  }
  __syncthreads();

  // ---- stage 3: softmax over the 2048-key rows, in LDS ----
  {
    int r = tid & 15, cs = tid >> 4;
    float m = -3.4e38f;
    for (int c = cs; c < NSEQ; c += 16) m = fmaxf(m, Sc[r * SS + c]);
    Red[r * 16 + cs] = m;
  }
  __syncthreads();
  if (tid < 16) {
    float m = Red[tid * 16];
    for (int i = 1; i < 16; ++i) m = fmaxf(m, Red[tid * 16 + i]);
    RowM[tid] = m;
  }
  __syncthreads();
  {
    int r = tid & 15, cs = tid >> 4;
    float m = RowM[r], s = 0.f;
    for (int c = cs; c < NSEQ; c += 16) {
      float e = __expf(Sc[r * SS + c] - m);
      Sc[r * SS + c] = e;
      s += e;
    }
    Red[r * 16 + cs] = s;
  }
  __syncthreads();
  if (tid < 16) {
    float s = 0.f;
    for (int i = 0; i < 16; ++i) s += Red[tid * 16 + i];
    RowS[tid] = 1.0f / s;
  }
  __syncthreads();

  float* arow = attn + ((size_t)(b * HEADS + h) * NSEQ + q0) * NSEQ;
#if HAVE_ASYNC_LDS
  // normalize in LDS, then stream attn straight from LDS to HBM (ASYNCcnt).
  for (int i = tid; i < 16 * NSEQ; i += 256) {
    int r = i >> 11, c = i & (NSEQ - 1);
    Sc[r * SS + c] *= RowS[r];
  }
  __syncthreads();
  for (int i = tid; i < (16 * NSEQ) / 4; i += 256) {
    int c4 = i * 4;
    int r = c4 >> 11, c = c4 & (NSEQ - 1);
    ASYNC_L2G(arow + (size_t)r * NSEQ + c, Sc + r * SS + c);
  }
  // stage 4 only READS Sc; async stores drain before Sc is overwritten below.
#else
  for (int i = tid; i < 16 * NSEQ; i += 256) {
    int r = i >> 11, c = i & (NSEQ - 1);
    float p = Sc[r * SS + c] * RowS[r];
    Sc[r * SS + c] = p;
    __builtin_nontemporal_store(p, &arow[(size_t)r * NSEQ + c]);
  }
  __syncthreads();
#endif

  // ---- stage 4: out = P @ V, wave owns its own 256-key strip of P ----
  const float* vbase = qkv + qkv_b + 2 * INNER + h * DH;  // V[key][f]
  v8f o[4] = {};
  const int kw0 = wave * 256;
  for (int kk = 0; kk < 256; kk += 4) {
    const int kg = kw0 + kk;
    v2f a;
    a.x = Sc[l16 * SS + kg + koff];
    a.y = Sc[l16 * SS + kg + koff + 1];
#pragma unroll
    for (int t = 0; t < 4; ++t) {
      v2f bb;
      bb.x = vbase[(size_t)(kg + koff) * QKVW + t * 16 + l16];
      bb.y = vbase[(size_t)(kg + koff + 1) * QKVW + t * 16 + l16];
      o[t] = wmma4(a, bb, o[t]);
    }
  }
#if HAVE_ASYNC_LDS
  wait_async0();            // attn stream out of Sc complete (this wave)
#endif
  __syncthreads();          // everyone done reading Sc -> reuse as reduce buf
  {
    float* pr = Sc + wave * 1024;      // [16][64] per-wave partial
#pragma unroll
    for (int t = 0; t < 4; ++t)
#pragma unroll
      for (int j = 0; j < 8; ++j) {
        int r = half ? (j + 8) : j;
        pr[r * 64 + t * 16 + l16] = o[t][j];
      }
  }
  __syncthreads();
  for (int i = tid; i < 16 * DH; i += 256) {
    int r = i >> 6, f = i & 63;
    float s = 0.f;
#pragma unroll
    for (int wv = 0; wv < 8; ++wv) s += Sc[wv * 1024 + r * 64 + f];
    aout[((size_t)b * NSEQ + q0 + r) * INNER + h * DH + f] = s;
  }
}

// =====================================================================
// Kernel 3: y = aout @ w_unify + b_unify ; LayerNorm(y) -> out
// Block owns 16 full rows (16x512) so LN fuses after the GEMM.
// grid = 512
// =====================================================================
__global__ void __launch_bounds__(256)
unify_ln_kernel(const float* __restrict__ aout, const float* __restrict__ w,
                const float* __restrict__ bias, const float* __restrict__ gamma,
                const float* __restrict__ beta, float* __restrict__ out) {
  __shared__ float Xs[16][DMODEL + 4];   // input tile, then reused as Y tile
  __shared__ float RedS[16][16];
  __shared__ float RedQ[16][16];
  __shared__ float Mu[16], Rs[16];

  const int row0 = blockIdx.x * 16;
  const int tid = threadIdx.x;
  const int wave = tid >> 5, lane = tid & 31;
  const int half = lane >> 4, l16 = lane & 15, koff = half * 2;

#if HAVE_ASYNC_LDS
  for (int i = tid; i < (16 * INNER) / 4; i += 256) {
    int c4 = i * 4;
    int r = c4 >> 9, c = c4 & (INNER - 1);
    ASYNC_G2L(aout + (size_t)(row0 + r) * INNER + c, &Xs[r][c]);
  }
  wait_async0();
#else
  for (int i = tid; i < 16 * INNER; i += 256) {
    int r = i >> 9, c = i & (INNER - 1);
    Xs[r][c] = aout[(size_t)(row0 + r) * INNER + c];
  }
#endif
  __syncthreads();

  v8f acc[4] = {};
  const int wcol0 = wave * 64;
  for (int kk = 0; kk < INNER; kk += 4) {
    v2f a;
    a.x = Xs[l16][kk + koff];
    a.y = Xs[l16][kk + koff + 1];
#pragma unroll
    for (int t = 0; t < 4; ++t) {
      int col = wcol0 + t * 16 + l16;
      v2f b;
      b.x = w[(size_t)(kk + koff) * DMODEL + col];
      b.y = w[(size_t)(kk + koff + 1) * DMODEL + col];
      acc[t] = wmma4(a, b, acc[t]);
    }
  }
  __syncthreads();                      // done reading Xs as input
#pragma unroll
  for (int t = 0; t < 4; ++t)
#pragma unroll
    for (int j = 0; j < 8; ++j) {
      int r = half ? (j + 8) : j;
      int c = wcol0 + t * 16 + l16;
      Xs[r][c] = acc[t][j] + bias[c];   // Y tile
    }
  __syncthreads();

  // LayerNorm: 16 threads per row, population variance
  {
    int r = tid & 15, cs = tid >> 4;
    float s = 0.f, s2 = 0.f;
    for (int c = cs; c < DMODEL; c += 16) {
      float v = Xs[r][c];
      s += v; s2 += v * v;
    }
    RedS[r][cs] = s; RedQ[r][cs] = s2;
  }
  __syncthreads();
  if (tid < 16) {
    float s = 0.f, s2 = 0.f;
    for (int i = 0; i < 16; ++i) { s += RedS[tid][i]; s2 += RedQ[tid][i]; }
    float mu = s * (1.0f / DMODEL);
    float var = s2 * (1.0f / DMODEL) - mu * mu;
    Mu[tid] = mu;
    Rs[tid] = rsqrtf(var + LN_EPS);
  }
  __syncthreads();
  for (int i = tid; i < 16 * DMODEL; i += 256) {
    int r = i >> 9, c = i & (DMODEL - 1);
    out[(size_t)(row0 + r) * DMODEL + c] =
        (Xs[r][c] - Mu[r]) * Rs[r] * gamma[c] + beta[c];
  }
}

// =====================================================================
extern "C" void kernel_launch(void* const* d_in, const int* in_sizes, int n_in,
                              void* d_out, int out_size, void* d_ws, size_t ws_size,
                              hipStream_t stream) {
  const float* x        = (const float*)d_in[0];
  const float* w_qkv    = (const float*)d_in[1];
  const float* w_unify  = (const float*)d_in[2];
  const float* b_unify  = (const float*)d_in[3];
  const float* ln_gamma = (const float*)d_in[4];
  const float* ln_beta  = (const float*)d_in[5];

  float* ln_out = (float*)d_out;                                   // [4,2048,512]
  float* attn   = ln_out + (size_t)BATCH * NSEQ * DMODEL;          // [4,8,2048,2048]

  float* qkv  = (float*)d_ws;                                      // [8192,1536]
  float* aout = qkv + (size_t)ROWS * QKVW;                         // [8192,512]

  qkv_gemm_kernel<<<dim3(ROWS / 16, 3), 256, 0, stream>>>(x, w_qkv, qkv);

  size_t smem2 = (size_t)(16 * SS + 16 * DH + 256 + 16 + 16) * sizeof(float);
  attention_kernel<<<BATCH * HEADS * (NSEQ / 16), 256, smem2, stream>>>(qkv, attn, aout);

  unify_ln_kernel<<<ROWS / 16, 256, 0, stream>>>(aout, w_unify, b_unify,
                                                 ln_gamma, ln_beta, ln_out);
}